// DyGraphConv2d_22333829939361
// MI455X (gfx1250) — compile-verified
//
#include <hip/hip_runtime.h>
#include <hip/hip_bf16.h>

// DyGraphConv2d (max-relative graph conv + 1x1 conv) for MI455X (gfx1250).
//
// Roofline: GEMM is 2.4 GFLOP (trivial); HBM traffic ~75MB (~3.2us @ 23.3TB/s);
// the 402MB of random gathers hit the 192MB L2 (feats = 25MB). => bandwidth
// bound, keep fp32 end-to-end and use V_WMMA_F32_16X16X4_F32 for the GEMM.

typedef __attribute__((ext_vector_type(2))) float v2f;
typedef __attribute__((ext_vector_type(8))) float v8f;

#define BB   16
#define CC   96
#define NN   4096              // H*W
#define KNN  16                // neighbors
#define BNN  (BB * NN)         // 65536 nodes
#define EE   (BNN * KNN)       // 1048576 edges
#define KDIM 192               // 2*C (interleaved x/agg channels)
#define LDS_STRIDE 193         // odd stride -> conflict-free LDS rows

// ---------------- Kernel 1: x [B,C,N] -> feats [B*N, C] ----------------
__global__ __launch_bounds__(1024) void transpose_kernel(
    const float* __restrict__ x, float* __restrict__ feats) {
  __shared__ float tile[32][33];
  const int n0 = blockIdx.x * 32;
  const int c0 = blockIdx.y * 32;
  const int b  = blockIdx.z;
  const int tx = threadIdx.x, ty = threadIdx.y;
  tile[ty][tx] = x[((size_t)b * CC + c0 + ty) * NN + n0 + tx];
  __syncthreads();
  feats[((size_t)b * NN + n0 + ty) * CC + c0 + tx] = tile[tx][ty];
}

// ---- Kernel 2: fused gather/max-relative + interleave + WMMA GEMM ----
// 256 threads = 8 waves; 64 nodes per block.
__global__ __launch_bounds__(256) void mrconv_kernel(
    const float* __restrict__ feats,      // [BN, 96]
    const long long* __restrict__ edge_src, // src row of edge_index, [E]
    const float* __restrict__ Wc,         // [96, 192]
    const float* __restrict__ bc,         // [96]
    float* __restrict__ out)              // [B, 96, N]
{
  __shared__ float lds_cat[64 * LDS_STRIDE];   // 64 nodes x 192 interleaved ch
  const int nodebase = blockIdx.x * 64;
  const int t = threadIdx.x;

  // ---------- Phase A: gather + max aggregate (4 threads/node) ----------
  {
    const int nd = t >> 2;             // local node 0..63
    const int cb = (t & 3) * 24;       // channel sub-range [cb, cb+24)
    const int node = nodebase + nd;
    const float4* fown = (const float4*)(feats + (size_t)node * CC + cb);
    float4 own[6];
    #pragma unroll
    for (int j = 0; j < 6; ++j) own[j] = fown[j];

    const float ninf = -__builtin_inff();
    float4 mx[6];
    #pragma unroll
    for (int j = 0; j < 6; ++j) mx[j] = make_float4(ninf, ninf, ninf, ninf);

    const long long* srcs = edge_src + (size_t)node * KNN;
    #pragma unroll 4
    for (int k = 0; k < KNN; ++k) {
      const long long s = srcs[k];
      const float4* fs = (const float4*)(feats + (size_t)s * CC + cb);
      #pragma unroll
      for (int j = 0; j < 6; ++j) {
        float4 v = fs[j];
        mx[j].x = fmaxf(mx[j].x, v.x);
        mx[j].y = fmaxf(mx[j].y, v.y);
        mx[j].z = fmaxf(mx[j].z, v.z);
        mx[j].w = fmaxf(mx[j].w, v.w);
      }
    }
    // interleave: cat[2c] = x_c, cat[2c+1] = max_k(f_src) - x_c
    float* row = lds_cat + nd * LDS_STRIDE + 2 * cb;
    #pragma unroll
    for (int j = 0; j < 6; ++j) {
      row[8*j + 0] = own[j].x;  row[8*j + 1] = mx[j].x - own[j].x;
      row[8*j + 2] = own[j].y;  row[8*j + 3] = mx[j].y - own[j].y;
      row[8*j + 4] = own[j].z;  row[8*j + 5] = mx[j].z - own[j].z;
      row[8*j + 6] = own[j].w;  row[8*j + 7] = mx[j].w - own[j].w;
    }
  }
  __syncthreads();

  // ---------- Phase B: WMMA f32 16x16x4 GEMM, bias, ReLU ----------
  const int lane  = t & 31;
  const int w     = t >> 5;            // wave 0..7
  const int m0    = (w >> 1) * 16;     // 16-node m tile
  const int nhalf = w & 1;             // output cols [nhalf*48, +48)
  const int l15   = lane & 15;
  const int lhi   = lane >> 4;         // 0/1: selects K pair within 16x4 A
  const int col0  = nhalf * 48 + l15;

  // A layout (32-bit 16x4): lane&15 = M; VGPR0/1 = K={0,1} (lanes<16) or
  // K={2,3} (lanes>=16).  B layout (4x16): lane&15 = N, same K-pair split.
  // B[k][n] = Wc[n][k] (Wc columns are already interleaved-ordered), so each
  // lane's {B[k],B[k+1]} is one aligned float2 load from Wc.
  const float* arow  = lds_cat + (m0 + l15) * LDS_STRIDE;
  const float* wrow0 = Wc + (size_t)col0 * KDIM;

  v8f acc0 = {}; v8f acc1 = {}; v8f acc2 = {};
  #pragma unroll 8
  for (int kb = 0; kb < KDIM; kb += 4) {
    const int kcol = kb + (lhi << 1);
    v2f a;
    a.x = arow[kcol];
    a.y = arow[kcol + 1];
    float2 w0 = *(const float2*)(wrow0 + kcol);
    float2 w1 = *(const float2*)(wrow0 + 16 * KDIM + kcol);
    float2 w2 = *(const float2*)(wrow0 + 32 * KDIM + kcol);
    v2f b0; b0.x = w0.x; b0.y = w0.y;
    v2f b1; b1.x = w1.x; b1.y = w1.y;
    v2f b2; b2.x = w2.x; b2.y = w2.y;
    acc0 = __builtin_amdgcn_wmma_f32_16x16x4_f32(false, a, false, b0,
                                                 (short)0, acc0, false, false);
    acc1 = __builtin_amdgcn_wmma_f32_16x16x4_f32(false, a, false, b1,
                                                 (short)0, acc1, false, false);
    acc2 = __builtin_amdgcn_wmma_f32_16x16x4_f32(false, a, false, b2,
                                                 (short)0, acc2, false, false);
  }

  // C/D layout: VGPR r holds M = r + 8*lhi, N = lane&15 -> the 8 acc elems
  // of one lane are 8 consecutive image positions => two float4 stores.
  const int b_idx = nodebase >> 12;                       // image index
  const int n_img = (nodebase & (NN - 1)) + m0 + (lhi << 3);
  #pragma unroll
  for (int i = 0; i < 3; ++i) {
    v8f acc = (i == 0) ? acc0 : (i == 1) ? acc1 : acc2;
    const int col = col0 + i * 16;
    const float bias = bc[col];
    float* op = out + ((size_t)b_idx * CC + col) * NN + n_img;
    float4 lo = make_float4(fmaxf(acc[0] + bias, 0.f), fmaxf(acc[1] + bias, 0.f),
                            fmaxf(acc[2] + bias, 0.f), fmaxf(acc[3] + bias, 0.f));
    float4 hi = make_float4(fmaxf(acc[4] + bias, 0.f), fmaxf(acc[5] + bias, 0.f),
                            fmaxf(acc[6] + bias, 0.f), fmaxf(acc[7] + bias, 0.f));
    ((float4*)op)[0] = lo;
    ((float4*)(op + 4))[0] = hi;
  }
}

extern "C" void kernel_launch(void* const* d_in, const int* in_sizes, int n_in,
                              void* d_out, int out_size, void* d_ws, size_t ws_size,
                              hipStream_t stream) {
  const float*     x    = (const float*)d_in[0];
  const long long* edge = (const long long*)d_in[1];  // [2, E] int64
  const float*     Wc   = (const float*)d_in[2];      // [96, 192]
  const float*     bc   = (const float*)d_in[3];      // [96]
  float* out   = (float*)d_out;
  float* feats = (float*)d_ws;                        // [BN, 96] = 25 MB

  dim3 tg(NN / 32, CC / 32, BB);
  dim3 tb(32, 32);
  transpose_kernel<<<tg, tb, 0, stream>>>(x, feats);

  // edge row 1 (src) starts at offset E
  mrconv_kernel<<<BNN / 64, 256, 0, stream>>>(feats, edge + EE, Wc, bc, out);
}